// GCN_Combined_59442347377119
// MI455X (gfx1250) — compile-verified
//
#include <hip/hip_runtime.h>
#include <math.h>

#define DIN  128
#define DH   128
#define DOUT 40
#define KD   128      // inner K dim of every GEMM in this net
#define LDSP 132      // padded LDS row stride (floats): lane->bank stride 4 mod 64, conflict-free

typedef __attribute__((ext_vector_type(2))) float v2f;
typedef __attribute__((ext_vector_type(8))) float v8f;
typedef __attribute__((ext_vector_type(4))) int   vi4;

// ---------------- degree / normalization ----------------

__global__ void k_deg_init(float* __restrict__ deg1, float* __restrict__ deg2, int n) {
  int i = blockIdx.x * blockDim.x + threadIdx.x;
  if (i < n) { deg1[i] = 1.0f; deg2[i] = 1.0f; }   // self-loop weight 1 in both graphs
}

__global__ void k_deg_accum(const int* __restrict__ dst, const float* __restrict__ ew,
                            float* __restrict__ deg1, float* __restrict__ deg2, int E) {
  int e = blockIdx.x * blockDim.x + threadIdx.x;
  if (e < E) {
    int d = dst[e];
    atomicAdd(&deg1[d], 1.0f);      // unit-weight graph (conv1, conv3)
    atomicAdd(&deg2[d], ew[e]);     // weighted graph (crf layer)
  }
}

__global__ void k_to_dinv(float* __restrict__ deg1, float* __restrict__ deg2, int n) {
  int i = blockIdx.x * blockDim.x + threadIdx.x;
  if (i < n) {                       // deg >= 1 always (self loop)
    deg1[i] = rsqrtf(deg1[i]);
    deg2[i] = rsqrtf(deg2[i]);
  }
}

// ---------------- fp32 WMMA GEMM: C[M,Nc] = A[M,128] @ B[128,Nc] ----------------
// One block (4 waves) per 16-row tile. The 16x128 A tile (2048 contiguous floats)
// is staged once into LDS via async global->LDS b128 (fallback: float4 copy),
// then all waves reuse it across their column tiles.
// Fragment layouts (V_WMMA_F32_16X16X4_F32):
//   A 16x4 : lanes 0-15 -> M=lane,    K=k,k+1 ; lanes 16-31 -> M=lane-16, K=k+2,k+3
//   B 4x16 : lanes 0-15 -> N=lane,    K=k,k+1 ; lanes 16-31 -> N=lane-16, K=k+2,k+3
//   C/D    : VGPR i, lanes 0-15 -> (M=i, N=lane) ; lanes 16-31 -> (M=i+8, N=lane-16)
// Requires M % 16 == 0 (true here: 50000 = 3125*16).

__global__ void k_gemm_wmma(const float* __restrict__ A, const float* __restrict__ B,
                            float* __restrict__ C, int M, int Nc) {
  __shared__ float As[16 * LDSP];
  const int lane   = threadIdx.x & 31;
  const int wave   = threadIdx.x >> 5;
  const int tm     = blockIdx.x;
  const int tilesN = (Nc + 15) >> 4;

  // ---- stage A row-block into LDS (global rows are contiguous; LDS rows padded) ----
  {
    const float* gsrc = A + (size_t)tm * 16 * KD;
#if __has_builtin(__builtin_amdgcn_global_load_async_to_lds_b128)
#pragma unroll
    for (int i = 0; i < 4; ++i) {
      int idx = (int)threadIdx.x + i * 128;        // float4 index within 16x128 tile
      int r = idx >> 5, c = (idx & 31) << 2;       // 32 float4 per row
      __attribute__((address_space(1))) vi4* gp =
          (__attribute__((address_space(1))) vi4*)(gsrc + r * KD + c);
      __attribute__((address_space(3))) vi4* lp =
          (__attribute__((address_space(3))) vi4*)&As[r * LDSP + c];
      __builtin_amdgcn_global_load_async_to_lds_b128(gp, lp, 0, 0);
    }
#if __has_builtin(__builtin_amdgcn_s_wait_asynccnt)
    __builtin_amdgcn_s_wait_asynccnt(0);
#else
    asm volatile("s_wait_asynccnt 0x0" ::: "memory");
#endif
#else
#pragma unroll
    for (int i = 0; i < 4; ++i) {
      int idx = (int)threadIdx.x + i * 128;
      int r = idx >> 5, c = (idx & 31) << 2;
      *(float4*)&As[r * LDSP + c] = *(const float4*)(gsrc + r * KD + c);
    }
#endif
  }
  __syncthreads();

  const int kk   = (lane >> 4) << 1;               // 0 for lanes 0-15, 2 for lanes 16-31
  const int arow = (lane & 15) * LDSP;             // A fragment row base in LDS

  for (int tn = wave; tn < tilesN; tn += 4) {      // wave-uniform -> EXEC all-1 for WMMA
    const int ncol    = tn * 16 + (lane & 15);
    const float bmask = (ncol < Nc) ? 1.0f : 0.0f; // branch-free column padding
    const int   ncolc = (ncol < Nc) ? ncol : (Nc - 1);

    v8f acc = {};
#pragma unroll 4
    for (int k = 0; k < KD; k += 4) {
      v2f av = *(const v2f*)&As[arow + k + kk];    // ds_load_b64, bank-conflict-free
      v2f bv;
      bv.x = B[(size_t)(k + kk)     * Nc + ncolc] * bmask;
      bv.y = B[(size_t)(k + kk + 1) * Nc + ncolc] * bmask;
      acc = __builtin_amdgcn_wmma_f32_16x16x4_f32(
          /*neg_a=*/false, av, /*neg_b=*/false, bv,
          /*c_mod=*/(short)0, acc, /*reuse_a=*/false, /*reuse_b=*/false);
    }

    const int rbase = tm * 16 + ((lane >> 4) << 3);
    if (ncol < Nc) {
#pragma unroll
      for (int i = 0; i < 8; ++i)
        C[(size_t)(rbase + i) * Nc + ncol] = acc[i];
    }
  }
}

// ---------------- aggregation ----------------

// agg[i,:] = H[i,:] * dinv[i]^2   (self-loop contribution, also zero-inits the buffer)
template <int F>
__global__ void k_init_agg(const float* __restrict__ H, const float* __restrict__ dinv,
                           float* __restrict__ agg, int total) {
  int tid = blockIdx.x * blockDim.x + threadIdx.x;
  if (tid >= total) return;
  int i = tid / F;
  float di = dinv[i];
  agg[tid] = H[tid] * di * di;
}

// per edge e: agg[dst[e],:] += H[src[e],:] * dinv[src]*w*dinv[dst]
// F/4 consecutive lanes cover one edge's feature row -> contiguous gather/atomic span
template <int F>
__global__ void k_scatter(const float* __restrict__ H, const int* __restrict__ src,
                          const int* __restrict__ dst, const float* __restrict__ ew,
                          const float* __restrict__ dinv, float* __restrict__ agg, int E) {
  const int F4 = F >> 2;
  int tid = blockIdx.x * blockDim.x + threadIdx.x;
  int e = tid / F4;
  if (e >= E) return;
  int c = (tid - e * F4) << 2;
  int s = src[e], d = dst[e];
  float w = ew ? ew[e] : 1.0f;
  float nrm = dinv[s] * w * dinv[d];
  const float4 v = *(const float4*)(H + (size_t)s * F + c);
  float* o = agg + (size_t)d * F + c;
  atomicAdd(o + 0, v.x * nrm);
  atomicAdd(o + 1, v.y * nrm);
  atomicAdd(o + 2, v.z * nrm);
  atomicAdd(o + 3, v.w * nrm);
}

// ---------------- epilogues ----------------

__global__ void k_epi_relu(const float* __restrict__ agg, const float* __restrict__ b,
                           float* __restrict__ out, int total) {
  int tid = blockIdx.x * blockDim.x + threadIdx.x;
  if (tid >= total) return;
  out[tid] = fmaxf(agg[tid] + b[tid & (DH - 1)], 0.0f);
}

// h (in place over h1):  h = ALPHA*h1 + (agg + b2)     [BETA = 1]
__global__ void k_epi_combine(const float* __restrict__ agg, const float* __restrict__ b,
                              float* __restrict__ h1, int total) {
  int tid = blockIdx.x * blockDim.x + threadIdx.x;
  if (tid >= total) return;
  h1[tid] = 0.1f * h1[tid] + agg[tid] + b[tid & (DH - 1)];
}

// logits = agg + b3 ; out = log_softmax(logits) row-wise. One wave per row (wave32).
__global__ void k_logsoftmax(const float* __restrict__ agg, const float* __restrict__ b,
                             float* __restrict__ out, int n) {
  int lane = threadIdx.x & 31;
  int wave = threadIdx.x >> 5;
  int row = blockIdx.x * (blockDim.x >> 5) + wave;
  if (row >= n) return;
  const float* a = agg + (size_t)row * DOUT;
  float v0 = a[lane] + b[lane];                               // lane 0..31 < 40, always valid
  bool  hi = (lane + 32) < DOUT;                              // lanes 0..7 own a second logit
  float v1 = hi ? (a[lane + 32] + b[lane + 32]) : v0;         // max-neutral filler
  float m = fmaxf(v0, v1);
#pragma unroll
  for (int off = 16; off > 0; off >>= 1) m = fmaxf(m, __shfl_xor(m, off, 32));
  float s = expf(v0 - m) + (hi ? expf(v1 - m) : 0.0f);
#pragma unroll
  for (int off = 16; off > 0; off >>= 1) s += __shfl_xor(s, off, 32);
  float lse = m + logf(s);
  float* o = out + (size_t)row * DOUT;
  o[lane] = v0 - lse;
  if (hi) o[lane + 32] = v1 - lse;
}

// ---------------- launch ----------------

extern "C" void kernel_launch(void* const* d_in, const int* in_sizes, int n_in,
                              void* d_out, int out_size, void* d_ws, size_t ws_size,
                              hipStream_t stream) {
  const float* x  = (const float*)d_in[0];
  const int*   ei = (const int*)d_in[1];     // [2,E] flat: src then dst
  const float* ew = (const float*)d_in[2];
  const float* W1 = (const float*)d_in[3];
  const float* b1 = (const float*)d_in[4];
  const float* W2 = (const float*)d_in[5];
  const float* b2 = (const float*)d_in[6];
  const float* W3 = (const float*)d_in[7];
  const float* b3 = (const float*)d_in[8];

  const int n = in_sizes[0] / DIN;           // 50000
  const int E = in_sizes[2];                 // 800000
  const int* src = ei;
  const int* dst = ei + E;

  // workspace layout (floats): tmp[n*128] | agg[n*128] | h1[n*128] | deg1[n] | deg2[n]
  float* ws   = (float*)d_ws;
  float* tmp  = ws;
  float* agg  = tmp + (size_t)n * DH;
  float* h1   = agg + (size_t)n * DH;
  float* deg1 = h1  + (size_t)n * DH;
  float* deg2 = deg1 + n;

  const int B = 256;
  const int tilesM = (n + 15) / 16;          // 3125 (n % 16 == 0)

  // degrees -> dinv (in place)
  k_deg_init <<<(n + B - 1) / B, B, 0, stream>>>(deg1, deg2, n);
  k_deg_accum<<<(E + B - 1) / B, B, 0, stream>>>(dst, ew, deg1, deg2, E);
  k_to_dinv  <<<(n + B - 1) / B, B, 0, stream>>>(deg1, deg2, n);

  const int totF = n * DH;                   // 6.4M
  const int totO = n * DOUT;                 // 2.0M

  // ---- layer 1: h1 = relu(gcn(x, unit weights, W1, b1)) ----
  {
    k_gemm_wmma<<<tilesM, 128, 0, stream>>>(x, W1, tmp, n, DH);
    k_init_agg<DH><<<(totF + B - 1) / B, B, 0, stream>>>(tmp, deg1, agg, totF);
    long sc = (long)E * (DH / 4);
    k_scatter<DH><<<(int)((sc + B - 1) / B), B, 0, stream>>>(tmp, src, dst, nullptr, deg1, agg, E);
    k_epi_relu<<<(totF + B - 1) / B, B, 0, stream>>>(agg, b1, h1, totF);
  }

  // ---- layer 2 (crf): h = 0.1*h1 + gcn(h1, edge_weight, W2, b2)  (in place over h1) ----
  {
    k_gemm_wmma<<<tilesM, 128, 0, stream>>>(h1, W2, tmp, n, DH);
    k_init_agg<DH><<<(totF + B - 1) / B, B, 0, stream>>>(tmp, deg2, agg, totF);
    long sc = (long)E * (DH / 4);
    k_scatter<DH><<<(int)((sc + B - 1) / B), B, 0, stream>>>(tmp, src, dst, ew, deg2, agg, E);
    k_epi_combine<<<(totF + B - 1) / B, B, 0, stream>>>(agg, b2, h1, totF);
  }

  // ---- layer 3: out = log_softmax(gcn(h, unit weights, W3, b3)) ----
  {
    k_gemm_wmma<<<tilesM, 128, 0, stream>>>(h1, W3, tmp, n, DOUT);
    k_init_agg<DOUT><<<(totO + B - 1) / B, B, 0, stream>>>(tmp, deg1, agg, totO);
    long sc = (long)E * (DOUT / 4);
    k_scatter<DOUT><<<(int)((sc + B - 1) / B), B, 0, stream>>>(tmp, src, dst, nullptr, deg1, agg, E);
    k_logsoftmax<<<(n + 7) / 8, 256, 0, stream>>>(agg, b3, (float*)d_out, n);
  }
}